// HierarchicalSoftmax_40785009443157
// MI455X (gfx1250) — compile-verified
//
#include <hip/hip_runtime.h>
#include <hip/hip_bf16.h>
#include <cstdint>

#define BATCH   1024
#define NHID    1024
#define NCLS    224
#define TPC     225
#define KC      64     // K-chunk staged through LDS per iteration

typedef float v2f __attribute__((ext_vector_type(2)));
typedef float v8f __attribute__((ext_vector_type(8)));

// -------------------- utility: zero a range of ints --------------------
__global__ void k_zero(int* __restrict__ p, int n) {
  int i = blockIdx.x * blockDim.x + threadIdx.x;
  if (i < n) p[i] = 0;
}

// -------------------- top GEMM: (1024x1024) @ (1024x224) + b --------------------
// One wave (32 lanes) per 16x16 output tile, fp32 WMMA 16x16x4, LDS staging.
__global__ __launch_bounds__(32)
void k_top_gemm(const float* __restrict__ X, const float* __restrict__ W,
                const float* __restrict__ bias, float* __restrict__ out) {
  const int n0 = blockIdx.x * 16;
  const int m0 = blockIdx.y * 16;
  const int lane = threadIdx.x;
  __shared__ float As[16][KC + 4];   // +4 pad: 16B-aligned rows, conflict-free column reads
  __shared__ float Bs[KC][16];
  v8f acc = {0.f, 0.f, 0.f, 0.f, 0.f, 0.f, 0.f, 0.f};
  const int mA = lane & 15;
  const int ko = (lane >> 4) << 1;   // 0 for lanes 0-15, 2 for lanes 16-31
  for (int k0 = 0; k0 < NHID; k0 += KC) {
    #pragma unroll
    for (int i = 0; i < 8; ++i) {          // 16x64 A tile, float4 coalesced
      int t  = lane + i * 32;
      int r  = t >> 4, c4 = (t & 15) << 2;
      float4 v = *(const float4*)(X + (size_t)(m0 + r) * NHID + k0 + c4);
      *(float4*)(&As[r][c4]) = v;
    }
    #pragma unroll
    for (int i = 0; i < 8; ++i) {          // 64x16 B tile, float4 coalesced
      int t  = lane + i * 32;
      int r  = t >> 2, c4 = (t & 3) << 2;
      float4 v = *(const float4*)(W + (size_t)(k0 + r) * NCLS + n0 + c4);
      *(float4*)(&Bs[r][c4]) = v;
    }
    __syncthreads();
    #pragma unroll
    for (int kk = 0; kk < KC; kk += 4) {
      v2f a, b;
      a.x = As[mA][kk + ko];     a.y = As[mA][kk + ko + 1];
      b.x = Bs[kk + ko][mA];     b.y = Bs[kk + ko + 1][mA];
      acc = __builtin_amdgcn_wmma_f32_16x16x4_f32(false, a, false, b,
                                                  (short)0, acc, false, false);
    }
    __syncthreads();
  }
  const int n   = lane & 15;
  const int mhi = (lane >> 4) << 3;
  const float bv = bias[n0 + n];
  #pragma unroll
  for (int r = 0; r < 8; ++r)
    out[(size_t)(m0 + mhi + r) * NCLS + n0 + n] = acc[r] + bv;
}

// -------------------- block reductions (blockDim == 256) --------------------
__device__ __forceinline__ void blk_max_argmax(float* rv, int* ri, int t,
                                               float v, int idx,
                                               float& smax, int& sidx) {
  rv[t] = v; ri[t] = idx;
  __syncthreads();
  for (int s = 128; s > 0; s >>= 1) {
    if (t < s) {
      float o = rv[t + s]; int oi = ri[t + s];
      if (o > rv[t] || (o == rv[t] && oi < ri[t])) { rv[t] = o; ri[t] = oi; }
    }
    __syncthreads();
  }
  smax = rv[0]; sidx = ri[0];
  __syncthreads();
}

__device__ __forceinline__ float blk_sum(float* rv, int t, float v) {
  rv[t] = v;
  __syncthreads();
  for (int s = 128; s > 0; s >>= 1) {
    if (t < s) rv[t] += rv[t + s];
    __syncthreads();
  }
  float r = rv[0];
  __syncthreads();
  return r;
}

// -------------------- top log-softmax + argmax + class decode --------------------
__global__ __launch_bounds__(256)
void k_top_softmax(const float* __restrict__ logits, const int* __restrict__ labels,
                   float* __restrict__ o_toplogp, float* __restrict__ o_topidx,
                   float* __restrict__ top_probs, float* __restrict__ target_top,
                   int* __restrict__ clsL, int* __restrict__ tokA,
                   int* __restrict__ clsP, int* __restrict__ cntL,
                   int* __restrict__ cntP) {
  const int b = blockIdx.x, t = threadIdx.x;
  const float* row = logits + (size_t)b * NCLS;
  __shared__ float rv[256];
  __shared__ int   ri[256];
  float v = (t < NCLS) ? row[t] : -3.402823466e38f;
  float smax; int sidx;
  blk_max_argmax(rv, ri, t, v, (t < NCLS) ? t : (NCLS - 1), smax, sidx);
  float e = (t < NCLS) ? __expf(v - smax) : 0.f;
  float ssum = blk_sum(rv, t, e);
  float logZ = __logf(ssum);
  if (t < NCLS) {
    o_toplogp[(size_t)b * NCLS + t] = v - smax - logZ;
    top_probs[(size_t)b * NCLS + t] = e / ssum;
  }
  if (t == 0) {
    o_topidx[b] = (float)sidx;
    clsP[b] = sidx;
    atomicAdd(&cntP[sidx], 1);
    int lab = labels[b];
    int cl  = lab / TPC;
    clsL[b] = cl;
    tokA[b] = lab - cl * TPC;
    atomicAdd(&cntL[cl], 1);
    target_top[b] = row[cl] - smax - logZ;
  }
}

// -------------------- per-class prefix scan (224 entries, trivial) --------------------
__global__ void k_prefix(const int* __restrict__ cntL, const int* __restrict__ cntP,
                         int* __restrict__ offL, int* __restrict__ offP,
                         int* __restrict__ curL, int* __restrict__ curP) {
  if (threadIdx.x == 0) {
    int s = 0;
    for (int i = 0; i < NCLS; ++i) { offL[i] = s; curL[i] = s; s += cntL[i]; }
  }
  if (threadIdx.x == 1) {
    int s = 0;
    for (int i = 0; i < NCLS; ++i) { offP[i] = s; curP[i] = s; s += cntP[i]; }
  }
}

// -------------------- scatter samples into per-class lists --------------------
__global__ __launch_bounds__(256)
void k_scatter(const int* __restrict__ clsL, const int* __restrict__ clsP,
               int* __restrict__ curL, int* __restrict__ curP,
               int* __restrict__ listL, int* __restrict__ listP) {
  int b = blockIdx.x * blockDim.x + threadIdx.x;
  if (b < BATCH) {
    int p = atomicAdd(&curL[clsL[b]], 1); listL[p] = b;
    int q = atomicAdd(&curP[clsP[b]], 1); listP[q] = b;
  }
}

// -------------------- class-batched bottom GEMM (the hotspot) --------------------
// grid = (15 N-tiles, 224 classes); one wave per block. Samples of each class
// form the M dimension (chunks of 16) so each class's 0.92MB weight slab is
// streamed once per N-tile instead of once per sample.
__global__ __launch_bounds__(32)
void k_bottom_gemm(const float* __restrict__ X, const float* __restrict__ Wb,
                   const float* __restrict__ bb,
                   const int* __restrict__ counts, const int* __restrict__ offs,
                   const int* __restrict__ list, float* __restrict__ out) {
  const int c   = blockIdx.y;
  const int cnt = counts[c];
  if (cnt == 0) return;
  const int n0  = blockIdx.x * 16;
  const int off = offs[c];
  const float* Wc = Wb + (size_t)c * NHID * TPC;
  const int lane = threadIdx.x;
  __shared__ float As[16][KC + 4];
  __shared__ float Bs[KC][16];
  __shared__ int   rows[16];
  const int mA = lane & 15;
  const int ko = (lane >> 4) << 1;
  for (int m0 = 0; m0 < cnt; m0 += 16) {
    const int mc = min(16, cnt - m0);
    if (lane < 16) rows[lane] = list[off + ((lane < mc) ? (m0 + lane) : m0)];
    __syncthreads();
    v8f acc = {0.f, 0.f, 0.f, 0.f, 0.f, 0.f, 0.f, 0.f};
    for (int k0 = 0; k0 < NHID; k0 += KC) {
      #pragma unroll
      for (int i = 0; i < 8; ++i) {          // gather 16 sample rows, 64 cols
        int t = lane + i * 32;
        int r = t >> 4, c4 = (t & 15) << 2;
        float4 v = *(const float4*)(X + (size_t)rows[r] * NHID + k0 + c4);
        *(float4*)(&As[r][c4]) = v;
      }
      #pragma unroll
      for (int i = 0; i < 32; ++i) {         // 64x16 slab of Wc (TPC=225 is odd -> b32)
        int t  = lane + i * 32;
        int r  = t >> 4, cc = t & 15;
        int nn = n0 + cc;
        Bs[r][cc] = (nn < TPC) ? Wc[(size_t)(k0 + r) * TPC + nn] : 0.f;
      }
      __syncthreads();
      #pragma unroll
      for (int kk = 0; kk < KC; kk += 4) {
        v2f a, b;
        a.x = As[mA][kk + ko];     a.y = As[mA][kk + ko + 1];
        b.x = Bs[kk + ko][mA];     b.y = Bs[kk + ko + 1][mA];
        acc = __builtin_amdgcn_wmma_f32_16x16x4_f32(false, a, false, b,
                                                    (short)0, acc, false, false);
      }
      __syncthreads();
    }
    const int n   = n0 + mA;
    const int mhi = (lane >> 4) << 3;
    if (n < TPC) {
      const float bv = bb[(size_t)c * TPC + n];
      #pragma unroll
      for (int r = 0; r < 8; ++r) {
        int m = mhi + r;
        if (m < mc) out[(size_t)rows[m] * TPC + n] = acc[r] + bv;
      }
    }
    __syncthreads();
  }
}

// -------------------- bottom (label path): logp, bot_idx, real_idx, target --------------------
__global__ __launch_bounds__(256)
void k_bottom_final_label(const float* __restrict__ logits, const int* __restrict__ tokA,
                          const float* __restrict__ target_top,
                          const float* __restrict__ o_topidx,
                          float* __restrict__ o_botlogp, float* __restrict__ o_botidx,
                          float* __restrict__ o_realidx, float* __restrict__ o_target) {
  const int b = blockIdx.x, t = threadIdx.x;
  const float* row = logits + (size_t)b * TPC;
  __shared__ float rv[256];
  __shared__ int   ri[256];
  float v = (t < TPC) ? row[t] : -3.402823466e38f;
  float smax; int sidx;
  blk_max_argmax(rv, ri, t, v, (t < TPC) ? t : (TPC - 1), smax, sidx);
  float e = (t < TPC) ? __expf(v - smax) : 0.f;
  float ssum = blk_sum(rv, t, e);
  float logZ = __logf(ssum);
  if (t < TPC) o_botlogp[(size_t)b * TPC + t] = v - smax - logZ;
  if (t == 0) {
    float tb = row[tokA[b]] - smax - logZ;
    float tp = target_top[b] + tb;
    o_target[b]  = tp;
    o_botidx[b]  = (float)sidx;
    o_realidx[b] = o_topidx[b] * (float)TPC + (float)sidx;
  }
}

// -------------------- bottom (pred path): softmax probs only --------------------
__global__ __launch_bounds__(256)
void k_bottom_softmax_pred(const float* __restrict__ logits, float* __restrict__ probs) {
  const int b = blockIdx.x, t = threadIdx.x;
  const float* row = logits + (size_t)b * TPC;
  __shared__ float rv[256];
  __shared__ int   ri[256];
  float v = (t < TPC) ? row[t] : -3.402823466e38f;
  float smax; int sidx;
  blk_max_argmax(rv, ri, t, v, (t < TPC) ? t : (TPC - 1), smax, sidx);
  float e = (t < TPC) ? __expf(v - smax) : 0.f;
  float ssum = blk_sum(rv, t, e);
  if (t < TPC) probs[(size_t)b * TPC + t] = e / ssum;
}

// -------------------- deterministic loss reduction --------------------
__global__ __launch_bounds__(256)
void k_loss(const float* __restrict__ target, float* __restrict__ o_loss) {
  __shared__ float rv[256];
  int t = threadIdx.x;
  float s = 0.f;
  for (int i = t; i < BATCH; i += 256) s += target[i];
  rv[t] = s;
  __syncthreads();
  for (int k = 128; k > 0; k >>= 1) {
    if (t < k) rv[t] += rv[t + k];
    __syncthreads();
  }
  if (t == 0) *o_loss = -rv[0] / (float)BATCH;
}

// -------------------- preds outer product: 206MB of stores, float4 --------------------
__global__ __launch_bounds__(256)
void k_preds(const float* __restrict__ top_probs, const float* __restrict__ botPp,
             float* __restrict__ preds) {
  const int b = blockIdx.x;
  __shared__ float tp[NCLS];
  __shared__ float bp[TPC];
  for (int i = threadIdx.x; i < NCLS; i += 256) tp[i] = top_probs[(size_t)b * NCLS + i];
  for (int i = threadIdx.x; i < TPC;  i += 256) bp[i] = botPp[(size_t)b * TPC + i];
  __syncthreads();
  float4* outrow = (float4*)(preds + (size_t)b * (NCLS * TPC));
  const int n4 = (NCLS * TPC) / 4;   // 12600
  for (int q = threadIdx.x; q < n4; q += 256) {
    int i = q * 4;
    float4 o;
    { int c = (i    ) / TPC; o.x = tp[c] * bp[(i    ) - c * TPC]; }
    { int c = (i + 1) / TPC; o.y = tp[c] * bp[(i + 1) - c * TPC]; }
    { int c = (i + 2) / TPC; o.z = tp[c] * bp[(i + 2) - c * TPC]; }
    { int c = (i + 3) / TPC; o.w = tp[c] * bp[(i + 3) - c * TPC]; }
    outrow[q] = o;
  }
}

// -------------------- launch --------------------
extern "C" void kernel_launch(void* const* d_in, const int* in_sizes, int n_in,
                              void* d_out, int out_size, void* d_ws, size_t ws_size,
                              hipStream_t stream) {
  (void)in_sizes; (void)n_in; (void)out_size; (void)ws_size;
  const float* X      = (const float*)d_in[0];
  const int*   labels = (const int*)d_in[1];
  const float* topW   = (const float*)d_in[2];
  const float* topB   = (const float*)d_in[3];
  const float* botW   = (const float*)d_in[4];
  const float* botB   = (const float*)d_in[5];

  float* out       = (float*)d_out;
  float* o_loss    = out;
  float* o_target  = out + 1;
  float* o_toplogp = o_target + BATCH;
  float* o_botlogp = o_toplogp + (size_t)BATCH * NCLS;
  float* o_topidx  = o_botlogp + (size_t)BATCH * TPC;
  float* o_botidx  = o_topidx + BATCH;
  float* o_realidx = o_botidx + BATCH;
  float* o_preds   = o_realidx + BATCH;

  float* top_logits = (float*)d_ws;
  float* top_probs  = top_logits + (size_t)BATCH * NCLS;
  float* botL       = top_probs  + (size_t)BATCH * NCLS;
  float* botP       = botL  + (size_t)BATCH * TPC;
  float* botPp      = botP  + (size_t)BATCH * TPC;
  float* target_top = botPp + (size_t)BATCH * TPC;
  int* clsL  = (int*)(target_top + BATCH);
  int* tokA  = clsL  + BATCH;
  int* clsP  = tokA  + BATCH;
  int* listL = clsP  + BATCH;
  int* listP = listL + BATCH;
  int* offL  = listP + BATCH;
  int* offP  = offL + 256;
  int* curL  = offP + 256;
  int* curP  = curL + 256;
  int* cntL  = curP + 256;
  int* cntP  = cntL + 256;

  k_zero<<<2, 256, 0, stream>>>(cntL, 512);  // cntL + cntP
  k_top_gemm<<<dim3(NCLS / 16, BATCH / 16), 32, 0, stream>>>(X, topW, topB, top_logits);
  k_top_softmax<<<BATCH, 256, 0, stream>>>(top_logits, labels, o_toplogp, o_topidx,
                                           top_probs, target_top, clsL, tokA, clsP,
                                           cntL, cntP);
  k_prefix<<<1, 32, 0, stream>>>(cntL, cntP, offL, offP, curL, curP);
  k_scatter<<<BATCH / 256, 256, 0, stream>>>(clsL, clsP, curL, curP, listL, listP);
  k_bottom_gemm<<<dim3((TPC + 15) / 16, NCLS), 32, 0, stream>>>(X, botW, botB,
                                                                cntL, offL, listL, botL);
  k_bottom_gemm<<<dim3((TPC + 15) / 16, NCLS), 32, 0, stream>>>(X, botW, botB,
                                                                cntP, offP, listP, botP);
  k_bottom_final_label<<<BATCH, 256, 0, stream>>>(botL, tokA, target_top, o_topidx,
                                                  o_botlogp, o_botidx, o_realidx, o_target);
  k_bottom_softmax_pred<<<BATCH, 256, 0, stream>>>(botP, botPp);
  k_loss<<<1, 256, 0, stream>>>(o_target, o_loss);
  k_preds<<<BATCH, 256, 0, stream>>>(top_probs, botPp, o_preds);
}